// Operator_46050639347866
// MI455X (gfx1250) — compile-verified
//
#include <hip/hip_runtime.h>

typedef float v2f __attribute__((ext_vector_type(2)));
typedef float v8f __attribute__((ext_vector_type(8)));

#define NLc 16
#define NSZ 256
#define NEc 100
#define NTc 12
#define NSL 48          // B*P*T slices
#define RK  4096        // reduction length (x,c)
#define E2  (NSZ*NSZ)   // 65536

__device__ __forceinline__ float clip1(float v){ return fminf(fmaxf(v, -1.f), 1.f); }

// ---------------- zero proj scratch ----------------
__global__ void zeroKernel(float* p, int n){
  int i = blockIdx.x*blockDim.x + threadIdx.x;
  if (i < n) p[i] = 0.f;
}

// ---------------- transpose x[c][xx] -> xt[r = xx*16+c] ----------------
__global__ void xtKernel(const float* __restrict__ xin, float* __restrict__ xt){
  int s = blockIdx.x, tid = threadIdx.x;
  const float* src = xin + (size_t)s*RK;
  float* dst = xt + (size_t)s*RK;
  #pragma unroll
  for (int k = 0; k < 16; k++){
    int r = tid + 256*k;
    int c = r & 15, xx = r >> 4;
    dst[r] = src[c*NSZ + xx];
  }
}

// ---------------- 48 x GEMM: kvt[s][X][Y] = sum_r K[X,r] * Q[Y,r] * t_s[r] ----------------
#define BT  64
#define CH  32
#define LST 34   // padded LDS row stride (bank-conflict free)

__global__ __launch_bounds__(128) void kvtKernel(const float* __restrict__ Km,
                                                 const float* __restrict__ Qm,
                                                 const float* __restrict__ xt,
                                                 float* __restrict__ kvt){
  __shared__ float sA[BT*LST];
  __shared__ float sB[BT*LST];
  int bid = blockIdx.x;
  int s   = bid >> 4;
  int t16 = bid & 15;
  int X0 = (t16 >> 2)*BT, Y0 = (t16 & 3)*BT;
  int tid = threadIdx.x;
  int cg = tid & 7;          // float4 column group within chunk
  int row0 = tid >> 3;       // 0..15
  int lane = tid & 31, wid = tid >> 5;
  int wM = wid >> 1, wN = wid & 1;
  int g = lane >> 4, m16 = lane & 15;
  const float* tvecs = xt + (size_t)s*RK;

  v8f acc[2][2] = {};
  int rA0 = (wM*32 +      m16)*LST;
  int rA1 = (wM*32 + 16 + m16)*LST;
  int rB0 = (wN*32 +      m16)*LST;
  int rB1 = (wN*32 + 16 + m16)*LST;

  for (int rb = 0; rb < RK; rb += CH){
    float4 tv = *(const float4*)&tvecs[rb + cg*4];
    #pragma unroll
    for (int it = 0; it < 4; it++){
      int row = row0 + 16*it;
      float4 a = *(const float4*)&Km[(size_t)(X0+row)*RK + rb + cg*4];
      float4 b = *(const float4*)&Qm[(size_t)(Y0+row)*RK + rb + cg*4];
      int o = row*LST + cg*4;
      sA[o+0] = a.x;      sA[o+1] = a.y;      sA[o+2] = a.z;      sA[o+3] = a.w;
      sB[o+0] = b.x*tv.x; sB[o+1] = b.y*tv.y; sB[o+2] = b.z*tv.z; sB[o+3] = b.w*tv.w;
    }
    if (rb + CH < RK){
      __builtin_prefetch(&Km[(size_t)(X0+row0)*RK + rb + CH], 0, 1);
      __builtin_prefetch(&Qm[(size_t)(Y0+row0)*RK + rb + CH], 0, 1);
    }
    __syncthreads();
    #pragma unroll
    for (int k = 0; k < CH; k += 4){
      int off = k + 2*g;
      v2f a0 = *(const v2f*)&sA[rA0 + off];
      v2f a1 = *(const v2f*)&sA[rA1 + off];
      v2f b0 = *(const v2f*)&sB[rB0 + off];
      v2f b1 = *(const v2f*)&sB[rB1 + off];
      acc[0][0] = __builtin_amdgcn_wmma_f32_16x16x4_f32(false, a0, false, b0, (short)0, acc[0][0], false, false);
      acc[0][1] = __builtin_amdgcn_wmma_f32_16x16x4_f32(false, a0, false, b1, (short)0, acc[0][1], false, false);
      acc[1][0] = __builtin_amdgcn_wmma_f32_16x16x4_f32(false, a1, false, b0, (short)0, acc[1][0], false, false);
      acc[1][1] = __builtin_amdgcn_wmma_f32_16x16x4_f32(false, a1, false, b1, (short)0, acc[1][1], false, false);
    }
    __syncthreads();
  }

  float* out = kvt + (size_t)s*E2;
  #pragma unroll
  for (int i = 0; i < 2; i++)
    #pragma unroll
    for (int j = 0; j < 2; j++){
      int Xb = X0 + wM*32 + i*16;
      int Yb = Y0 + wN*32 + j*16;
      #pragma unroll
      for (int v = 0; v < 8; v++){
        int m = v + 8*g;
        out[(size_t)(Xb+m)*NSZ + Yb + m16] = acc[i][j][v];
      }
    }
}

// ---------------- proj[ps][n] = sum_e clip(Ano)[e][n] * kvt[bp][J][e]  (atomic-combined) ----------------
#define ECH 1024
__global__ void projKernel(const float* __restrict__ Ano, const float* __restrict__ kvt,
                           float* __restrict__ proj){
  __shared__ float sred[256];
  int bid = blockIdx.x;
  int ps = bid >> 6;          // 0..43  (bp*11 + J)
  int chunk = bid & 63;
  int bp = ps / 11, J = ps % 11;
  const float* kv = kvt + (size_t)(bp*NTc + J)*E2 + (size_t)chunk*ECH;
  const float* an = Ano + (size_t)chunk*ECH*NEc;
  int tid = threadIdx.x;
  int n = tid & 127, half = tid >> 7;
  float acc = 0.f;
  if (n < NEc){
    for (int i = 0; i < ECH/2; i++){
      int e = half*(ECH/2) + i;
      acc += clip1(an[(size_t)e*NEc + n]) * kv[e];
    }
  }
  sred[tid] = acc;
  __syncthreads();
  if (half == 0 && n < NEc) atomicAdd(&proj[ps*NEc + n], acc + sred[tid + 128]);
}

// ---------------- WA[X][n] = sum_Y w[Y] * clip(Aon)[X][Y][n] ----------------
__global__ void waKernel(const float* __restrict__ Aon, const float* __restrict__ w,
                         float* __restrict__ WA){
  int X = blockIdx.x, n = threadIdx.x;
  if (n >= NEc) return;
  const float* a = Aon + (size_t)X*NSZ*NEc + n;
  float acc = 0.f;
  for (int y = 0; y < NSZ; y++) acc += w[y]*clip1(a[(size_t)y*NEc]);
  WA[X*NEc + n] = acc;
}

// ---------------- d[s][n] = proj[s][10][n] + sum_J A_seq[9-J]*proj[s][J][n] ----------------
__global__ void dKernel(const float* __restrict__ Ann, const float* __restrict__ proj,
                        float* __restrict__ dv){
  int n = threadIdx.x;
  if (n >= NEc) return;
  float ann = Ann[n];
  float aseq[10];
  float a = ann;
  #pragma unroll
  for (int j = 0; j < 10; j++){ aseq[j] = a; a = clip1(a*ann); }
  for (int s = 0; s < 4; s++){
    float s2 = 0.f;
    #pragma unroll
    for (int J = 0; J < 10; J++) s2 += aseq[9-J]*proj[(s*11 + J)*NEc + n];
    dv[s*NEc + n] = proj[(s*11 + 10)*NEc + n] + s2;
  }
}

// ---------------- final: hw[X] = e1 + WA·d ; y[s][c] = sum_X q[c][X]*hw[X] ----------------
__global__ __launch_bounds__(256) void finalKernel(const float* __restrict__ xin,
                                                   const float* __restrict__ V,
                                                   const float* __restrict__ Aoo,
                                                   const float* __restrict__ w,
                                                   const float* __restrict__ kvt,
                                                   const float* __restrict__ WA,
                                                   const float* __restrict__ dv,
                                                   float* __restrict__ out){
  __shared__ float sd[NEc];
  __shared__ float sw[NSZ];
  __shared__ float shw[NSZ];
  __shared__ float sred[NSZ];
  int s = blockIdx.x, X = threadIdx.x;
  if (X < NEc) sd[X] = dv[s*NEc + X];
  sw[X] = w[X];
  __syncthreads();

  const float* kvl = kvt + (size_t)(s*NTc + NTc-1)*E2 + (size_t)X*NSZ;
  const float* ao  = Aoo + (size_t)X*NSZ;
  float acc = 0.f;
  for (int y = 0; y < NSZ; y++) acc += sw[y]*ao[y]*kvl[y];
  const float* wax = WA + X*NEc;
  for (int n = 0; n < NEc; n++) acc += wax[n]*sd[n];
  shw[X] = acc;
  __syncthreads();

  const float* xl = xin + (size_t)(s*NTc + NTc-1)*RK;
  for (int c = 0; c < NLc; c++){
    const float* vr = V + (size_t)X*RK + c;
    const float* xc = xl + c*NSZ;
    float q = 0.f;
    for (int xx = 0; xx < NSZ; xx++) q += vr[(size_t)xx*NLc]*xc[xx];
    sred[X] = q*shw[X];
    __syncthreads();
    for (int off = 128; off >= 1; off >>= 1){
      if (X < off) sred[X] += sred[X + off];
      __syncthreads();
    }
    if (X == 0) out[s*NLc + c] = sred[0];
    __syncthreads();
  }
}

extern "C" void kernel_launch(void* const* d_in, const int* in_sizes, int n_in,
                              void* d_out, int out_size, void* d_ws, size_t ws_size,
                              hipStream_t stream){
  const float* x   = (const float*)d_in[0];
  const float* K   = (const float*)d_in[1];
  const float* Q   = (const float*)d_in[2];
  const float* V   = (const float*)d_in[3];
  const float* Aoo = (const float*)d_in[4];
  const float* Ann = (const float*)d_in[5];
  const float* Aon = (const float*)d_in[6];
  const float* Ano = (const float*)d_in[7];
  const float* w   = (const float*)d_in[8];
  float* out = (float*)d_out;

  float* ws  = (float*)d_ws;
  float* xt   = ws;                          // 48*4096
  float* kvt  = xt  + (size_t)NSL*RK;        // 48*65536
  float* proj = kvt + (size_t)NSL*E2;        // 44*100
  float* dv   = proj + 44*NEc;               // 4*100
  float* WA   = dv   + 4*NEc;                // 256*100

  zeroKernel<<<(44*NEc + 255)/256, 256, 0, stream>>>(proj, 44*NEc);
  xtKernel  <<<NSL, 256, 0, stream>>>(x, xt);
  kvtKernel <<<NSL*16, 128, 0, stream>>>(K, Q, xt, kvt);
  projKernel<<<44*64, 256, 0, stream>>>(Ano, kvt, proj);
  waKernel  <<<NSZ, 128, 0, stream>>>(Aon, w, WA);
  dKernel   <<<1, 128, 0, stream>>>(Ann, proj, dv);
  finalKernel<<<4, 256, 0, stream>>>(x, V, Aoo, w, kvt, WA, dv, out);
}